// Chroma_51788715655341
// MI455X (gfx1250) — compile-verified
//
#include <hip/hip_runtime.h>

typedef __attribute__((ext_vector_type(2))) float v2f;
typedef __attribute__((ext_vector_type(8))) float v8f;

#define T_LEN   160000
#define C_CH    48
#define KW      129
#define KW_PAD  132
#define WIN_L   400
#define HOP_L   160
#define NWIN    1000
#define CHUNK   80
#define NCHUNK  5
#define SEG     528      // WIN_L + KW - 1
#define SEG_PAD 536
#define NPAIR   1128     // 48*47/2
#define COV_STRIDE 49

__global__ __launch_bounds__(256)
void chroma_corr_kernel(const float* __restrict__ wav,
                        const float* __restrict__ W,
                        float* __restrict__ out)
{
    __shared__ float wavSeg[SEG_PAD];
    __shared__ float Wlds[C_CH * KW_PAD];
    __shared__ float featChunk[CHUNK * C_CH];
    __shared__ float sumsLds[C_CH];
    __shared__ float covLds[C_CH * COV_STRIDE];

    const int tid  = threadIdx.x;
    const int lane = tid & 31;
    const int wave = tid >> 5;
    const int half = lane >> 4;   // 0: lanes 0-15, 1: lanes 16-31
    const int l16  = lane & 15;

    const int blk = blockIdx.x;          // = b*NWIN + n
    const int b   = blk / NWIN;
    const int n   = blk % NWIN;
    const int s   = n * HOP_L;
    const int cnt = (T_LEN - s < WIN_L) ? (T_LEN - s) : WIN_L;
    const float invCnt = 1.0f / (float)cnt;

    // ---- load wav segment [s-64, s+463] with zero padding, tail zeroed ----
    for (int i = tid; i < SEG_PAD; i += 256) {
        int t = s - 64 + i;
        float v = 0.0f;
        if (i < SEG && t >= 0 && t < T_LEN) v = wav[b * T_LEN + t];
        wavSeg[i] = v;
    }
    // ---- load W padded to K=132 with zeros ----
    for (int i = tid; i < C_CH * KW_PAD; i += 256) {
        int c = i / KW_PAD;
        int k = i % KW_PAD;
        Wlds[i] = (k < KW) ? W[c * KW + k] : 0.0f;
    }
    __syncthreads();

    // symmetric Gram tile pair for waves 0..5: (0,0)(0,1)(0,2)(1,1)(1,2)(2,2)
    int ci = 0, cj = 0;
    if (wave < 6) {
        ci = (wave < 3) ? 0 : ((wave < 5) ? 1 : 2);
        cj = (wave < 3) ? wave : ((wave < 5) ? (wave - 2) : 2);
    }
    v8f gacc = {};          // persistent Gram accumulator (waves 0..5)
    float colSum = 0.0f;    // column-sum accumulator (threads 192..239)

    for (int chunk = 0; chunk < NCHUNK; ++chunk) {
        const int wb = chunk * CHUNK;

        // ===== conv phase: 15 tile jobs (5 M-tiles x 3 N-tiles), 8 waves =====
        for (int j = wave; j < 15; j += 8) {
            const int mt = j / 3;
            const int nt = j % 3;
            // A (16x4 f32): lane<16 -> K=0,1 ; lane>=16 -> K=2,3
            const int abase = wb + mt * 16 + l16 + half * 2;
            const float* wrow = &Wlds[(nt * 16 + l16) * KW_PAD + half * 2];
            v8f acc = {};
            for (int kt = 0; kt < 33; ++kt) {
                v2f av, bv;
                av.x = wavSeg[abase + kt * 4];
                av.y = wavSeg[abase + kt * 4 + 1];
                bv.x = wrow[kt * 4];
                bv.y = wrow[kt * 4 + 1];
                acc = __builtin_amdgcn_wmma_f32_16x16x4_f32(
                        false, av, false, bv, (short)0, acc, false, false);
            }
            // D (16x16 f32): vgpr r, lane<16 -> M=r ; lane>=16 -> M=r+8
            #pragma unroll
            for (int r = 0; r < 8; ++r) {
                int M   = r + 8 * half;
                int row = mt * 16 + M;                       // row within chunk
                float v = ((wb + row) < cnt) ? acc[r] : 0.0f; // tail mask
                featChunk[row * C_CH + nt * 16 + l16] = v;
            }
        }
        __syncthreads();

        // ===== Gram accumulation (waves 0-5) / column sums (waves 6-7) =====
        if (wave < 6) {
            const int ca = ci * 16 + l16;
            const int cb = cj * 16 + l16;
            for (int kt = 0; kt < CHUNK / 4; ++kt) {
                const int r0 = kt * 4 + half * 2;
                v2f av, bv;
                av.x = featChunk[r0 * C_CH + ca];
                av.y = featChunk[(r0 + 1) * C_CH + ca];
                bv.x = featChunk[r0 * C_CH + cb];
                bv.y = featChunk[(r0 + 1) * C_CH + cb];
                gacc = __builtin_amdgcn_wmma_f32_16x16x4_f32(
                         false, av, false, bv, (short)0, gacc, false, false);
            }
        } else if (tid >= 192 && tid < 192 + C_CH) {
            const int c = tid - 192;
            float sloc = 0.0f;
            for (int r = 0; r < CHUNK; ++r) sloc += featChunk[r * C_CH + c];
            colSum += sloc;
        }
        __syncthreads();
    }

    if (tid >= 192 && tid < 192 + C_CH) sumsLds[tid - 192] = colSum;
    __syncthreads();

    // ===== finalize: cov = (G - s*s^T/cnt) / (C-1) into LDS =====
    if (wave < 6) {
        #pragma unroll
        for (int r = 0; r < 8; ++r) {
            int M = r + 8 * half;
            int c = ci * 16 + M;
            int d = cj * 16 + l16;
            float cov = (gacc[r] - sumsLds[c] * sumsLds[d] * invCnt)
                        * (1.0f / 47.0f);
            covLds[c * COV_STRIDE + d] = cov;
        }
    }
    __syncthreads();

    // ===== correlation + triu(k=1) output =====
    for (int p = tid; p < NPAIR; p += 256) {
        int i = 0, rem = p;
        while (rem >= 47 - i) { rem -= 47 - i; ++i; }
        int jj = i + 1 + rem;
        float g  = covLds[i  * COV_STRIDE + jj];
        float di = covLds[i  * COV_STRIDE + i ];
        float dj = covLds[jj * COV_STRIDE + jj];
        float denom = sqrtf(di) * sqrtf(dj);
        if (g == 0.0f) g = 1e-10f;
        float cor = g / denom;
        // NaN-preserving clamp, then NaN -> 0 (matches reference order)
        cor = (cor >  1.0f) ?  1.0f : cor;
        cor = (cor < -1.0f) ? -1.0f : cor;
        if (cor != cor) cor = 0.0f;
        out[(size_t)blk * NPAIR + p] = cor;
    }
}

extern "C" void kernel_launch(void* const* d_in, const int* in_sizes, int n_in,
                              void* d_out, int out_size, void* d_ws, size_t ws_size,
                              hipStream_t stream) {
    (void)in_sizes; (void)n_in; (void)d_ws; (void)ws_size; (void)out_size;
    const float* wav = (const float*)d_in[0];   // (4, 160000) f32
    const float* W   = (const float*)d_in[1];   // (48, 1, 129) f32
    float* out = (float*)d_out;                 // (4, 1000, 1128) f32
    dim3 grid(4 * NWIN);
    dim3 block(256);
    chroma_corr_kernel<<<grid, block, 0, stream>>>(wav, W, out);
}